// MONForwardBackwardAnderson_25735444038087
// MI455X (gfx1250) — compile-verified
//
#include <hip/hip_runtime.h>
#include <hip/hip_bf16.h>

typedef float v2f __attribute__((ext_vector_type(2)));
typedef float v4f __attribute__((ext_vector_type(4)));
typedef float v8f __attribute__((ext_vector_type(8)));
typedef int   i4v __attribute__((vector_size(16)));   // matches builtin param type

#define AS1 __attribute__((address_space(1)))
#define AS3 __attribute__((address_space(3)))

#define BSZ     512
#define IN_DIM  1024
#define OUT_DIM 2048
#define M_AND   5
#define LAMREG  1e-4f
#define M_MON   1.0f
#define MAX_IT  50

enum { EPI_BIASVEC = 0, EPI_SKEW = 1, EPI_RELUB = 2 };

// ---------------------------------------------------------------------------
// CDNA5 async global->LDS copies (ASYNCcnt-tracked, bypass VGPRs).
// ---------------------------------------------------------------------------
#if __has_builtin(__builtin_amdgcn_global_load_async_to_lds_b128) && \
    __has_builtin(__builtin_amdgcn_global_load_async_to_lds_b32)
#define HAVE_ASYNC_LDS 1
#endif

__device__ __forceinline__ void cp_lds_b128(const float* g, float* l)
{
#ifdef HAVE_ASYNC_LDS
    __builtin_amdgcn_global_load_async_to_lds_b128(
        (AS1 i4v*)(uintptr_t)g,
        (AS3 i4v*)(uint32_t)(uintptr_t)l, 0, 0);
#else
    *(v4f*)l = *(const v4f*)g;
#endif
}

__device__ __forceinline__ void cp_lds_b32(const float* g, float* l)
{
#ifdef HAVE_ASYNC_LDS
    __builtin_amdgcn_global_load_async_to_lds_b32(
        (AS1 int*)(uintptr_t)g,
        (AS3 int*)(uint32_t)(uintptr_t)l, 0, 0);
#else
    *l = *g;
#endif
}

__device__ __forceinline__ void wait_async()
{
#ifdef HAVE_ASYNC_LDS
#if __has_builtin(__builtin_amdgcn_s_wait_asynccnt)
    __builtin_amdgcn_s_wait_asynccnt(0);
#else
    asm volatile("s_wait_asynccnt 0" ::: "memory");
#endif
#endif
}

// ---------------------------------------------------------------------------
// fp32 WMMA GEMM:  C[i,j] = epilogue( sum_k opA(i,k) * opB(k,j) )
// Block tile 128x64, 8 wave32 waves (4x2), wave tile 32x32 = 4 f32
// accumulators. K staged through DOUBLE-BUFFERED LDS in steps of 16 via
// async copies: per iteration  wait_async -> barrier -> issue next tile's
// asyncs into alternate buffer -> 16 WMMAs on current buffer.  The DMA of
// tile k+1 overlaps the matrix math of tile k; one barrier per K-block.
//
// LDS layouts:
//   As[2][128][20] row-major : pitch 80 B (16B-aligned rows for b128 async
//                              writes; 20r mod 64 -> conflict-free b64 reads)
//   Bs[2][8][160] K-pair-interleaved: element (k,j) at Bs[.][k>>1][2j+(k&1)]
//                              so a B fragment is ONE aligned ds_load_b64.
// Fragment layouts per ISA 7.12.2 (16x4 A: v0/v1 = K{0,1} lanes<16, K{2,3}
// lanes>=16; 4x16 B mirrored; C/D VGPR r = rows r / r+8).
// ---------------------------------------------------------------------------
template<int EPI, bool TA, bool TB>
__global__ __launch_bounds__(256)
void wmma_gemm(const float* __restrict__ A, int lda,
               const float* __restrict__ B, int ldb,
               float* __restrict__ C, int ldc,
               int Kdim,
               const float* __restrict__ aux, int auxld,
               float diagAdd)
{
    __shared__ float As[2][128][20];
    __shared__ float Bs[2][8][160];

    const int tid  = threadIdx.x;
    const int lane = tid & 31;
    const int wave = tid >> 5;
    const int wr   = (wave >> 1) * 32;   // 0,32,64,96
    const int wc   = (wave & 1) * 32;    // 0,32
    const int r16  = lane & 15;
    const int half = lane >> 4;

    const int gRow0 = blockIdx.y * 128;
    const int gCol0 = blockIdx.x * 64;

    // stage tile at K-offset k0 into buffer `buf`
    auto stage = [&](int buf, int k0) {
        if (!TA) {
            #pragma unroll
            for (int u = 0; u < 2; ++u) {
                int e = tid + 256 * u;
                int ai = e >> 2, av = (e & 3) * 4;
                cp_lds_b128(&A[(size_t)(gRow0 + ai) * lda + k0 + av], &As[buf][ai][av]);
            }
        } else {
            #pragma unroll
            for (int u = 0; u < 8; ++u) {
                int e = tid + 256 * u;
                int ai = e & 127, ak = e >> 7;      // coalesced along ai
                cp_lds_b32(&A[(size_t)(k0 + ak) * lda + gRow0 + ai], &As[buf][ai][ak]);
            }
        }
        #pragma unroll
        for (int u = 0; u < 4; ++u) {
            int e = tid + 256 * u;
            int bk = e >> 6, bj = e & 63;           // bj fastest -> coalesced (!TB)
            const float* g = TB ? &B[(size_t)(gCol0 + bj) * ldb + k0 + bk]
                                : &B[(size_t)(k0 + bk) * ldb + gCol0 + bj];
            cp_lds_b32(g, &Bs[buf][bk >> 1][bj * 2 + (bk & 1)]);
        }
    };

    v8f acc[2][2] = {};

    stage(0, 0);                         // prologue: tile 0 in flight
    int cur = 0;
    for (int k0 = 0; k0 < Kdim; k0 += 16) {
        wait_async();                    // my loads for buf[cur] done
        __syncthreads();                 // everyone's loads done + prev compute done
        if (k0 + 16 < Kdim)
            stage(cur ^ 1, k0 + 16);     // next tile's DMA overlaps this compute

        // ---- 4 K-steps of 4; 4 WMMAs per step, all from buf[cur] ----
        #pragma unroll
        for (int kk = 0; kk < 16; kk += 4) {
            v2f a0 = *(const v2f*)&As[cur][wr +      r16][kk + 2 * half];
            v2f a1 = *(const v2f*)&As[cur][wr + 16 + r16][kk + 2 * half];
            v2f b0 = *(const v2f*)&Bs[cur][(kk >> 1) + half][(wc +      r16) * 2];
            v2f b1 = *(const v2f*)&Bs[cur][(kk >> 1) + half][(wc + 16 + r16) * 2];
            acc[0][0] = __builtin_amdgcn_wmma_f32_16x16x4_f32(false, a0, false, b0,
                                                              (short)0, acc[0][0], false, false);
            acc[0][1] = __builtin_amdgcn_wmma_f32_16x16x4_f32(false, a0, false, b1,
                                                              (short)0, acc[0][1], false, false);
            acc[1][0] = __builtin_amdgcn_wmma_f32_16x16x4_f32(false, a1, false, b0,
                                                              (short)0, acc[1][0], false, false);
            acc[1][1] = __builtin_amdgcn_wmma_f32_16x16x4_f32(false, a1, false, b1,
                                                              (short)0, acc[1][1], false, false);
        }
        cur ^= 1;
    }

    // ---- epilogue + store ----
    #pragma unroll
    for (int rt = 0; rt < 2; ++rt)
        #pragma unroll
        for (int ct = 0; ct < 2; ++ct)
            #pragma unroll
            for (int r = 0; r < 8; ++r) {
                int gi = gRow0 + wr + rt * 16 + r + 8 * half;
                int gj = gCol0 + wc + ct * 16 + r16;
                float v = acc[rt][ct][r];
                if (EPI == EPI_BIASVEC) {            // + broadcast bias vector
                    v += aux[gj];
                } else if (EPI == EPI_SKEW) {        // M = -A^T A + B^T - B + d*I
                    v = -v + aux[(size_t)gj * auxld + gi]
                           - aux[(size_t)gi * auxld + gj]
                           + ((gi == gj) ? diagAdd : 0.f);
                } else {                             // relu(acc + bias matrix)
                    v = fmaxf(v + aux[(size_t)gi * auxld + gj], 0.f);
                }
                C[(size_t)gi * ldc + gj] = v;
            }
}

// ---------------------------------------------------------------------------
__global__ void fill_zero(float* p, long n)
{
    long i = (long)blockIdx.x * blockDim.x + threadIdx.x;
    long s = (long)gridDim.x * blockDim.x;
    for (; i < n; i += s) p[i] = 0.f;
}

// F[:,0] = relu(bias); Z[:,1] = relu(bias)
__global__ void bootstrap_f0(const float* __restrict__ bias,
                             float* __restrict__ F, float* __restrict__ Z)
{
    int i = blockIdx.x * blockDim.x + threadIdx.x;   // 512*2048
    int b = i >> 11, d = i & 2047;
    float f0 = fmaxf(bias[i], 0.f);
    F[((size_t)b * M_AND + 0) * OUT_DIM + d] = f0;
    Z[((size_t)b * M_AND + 1) * OUT_DIM + d] = f0;
}

// Per batch row: G = F - Z; 5x5 Gram via block reduction; Frobenius
// normalization; bordered 6x6 solve (Gaussian elim, partial pivoting since
// H[0,0] = 0) -> alpha[b,0..4].
__global__ __launch_bounds__(256)
void anderson_alpha(const float* __restrict__ Z, const float* __restrict__ F,
                    float* __restrict__ alpha, int it)
{
    __shared__ float red[256];
    __shared__ float psum[15];
    const int b = blockIdx.x, tid = threadIdx.x;

    float s[15];
    #pragma unroll
    for (int p = 0; p < 15; ++p) s[p] = 0.f;

    for (int d = tid; d < OUT_DIM; d += 256) {
        float g[M_AND];
        #pragma unroll
        for (int m = 0; m < M_AND; ++m) {
            size_t off = ((size_t)b * M_AND + m) * OUT_DIM + d;
            g[m] = F[off] - Z[off];
        }
        int p = 0;
        #pragma unroll
        for (int m = 0; m < M_AND; ++m)
            #pragma unroll
            for (int nn = m; nn < M_AND; ++nn)
                s[p++] += g[m] * g[nn];
    }
    for (int p = 0; p < 15; ++p) {
        red[tid] = s[p];
        __syncthreads();
        for (int off = 128; off > 0; off >>= 1) {
            if (tid < off) red[tid] += red[tid + off];
            __syncthreads();
        }
        if (tid == 0) psum[p] = red[0];
        __syncthreads();
    }

    if (tid == 0) {
        const int n = it < M_AND ? it : M_AND;
        float GG[M_AND][M_AND];
        int p = 0;
        for (int m = 0; m < M_AND; ++m)
            for (int nn = m; nn < M_AND; ++nn) {
                GG[m][nn] = psum[p]; GG[nn][m] = psum[p]; ++p;
            }
        float nrm = 0.f;
        for (int m = 0; m < M_AND; ++m)
            for (int nn = 0; nn < M_AND; ++nn) nrm += GG[m][nn] * GG[m][nn];
        const float inv = 1.f / (sqrtf(nrm) + 1e-6f);

        float Hm[6][7];
        for (int r = 0; r < 6; ++r)
            for (int c = 0; c < 7; ++c) Hm[r][c] = 0.f;
        for (int j = 0; j < M_AND; ++j) {
            float mk = (j < n) ? 1.f : 0.f;
            Hm[0][1 + j] = mk;
            Hm[1 + j][0] = mk;
        }
        for (int i = 0; i < M_AND; ++i)
            for (int j = 0; j < M_AND; ++j) {
                bool act = (i < n) && (j < n);
                Hm[1 + i][1 + j] = act ? (GG[i][j] * inv + (i == j ? LAMREG : 0.f))
                                       : (i == j ? 1.f : 0.f);
            }
        Hm[0][6] = 1.f;   // RHS e0

        for (int c = 0; c < 6; ++c) {
            int piv = c; float best = fabsf(Hm[c][c]);
            for (int r = c + 1; r < 6; ++r) {
                float v = fabsf(Hm[r][c]);
                if (v > best) { best = v; piv = r; }
            }
            if (piv != c)
                for (int cc = c; cc < 7; ++cc) {
                    float t = Hm[c][cc]; Hm[c][cc] = Hm[piv][cc]; Hm[piv][cc] = t;
                }
            float pv = Hm[c][c];
            for (int r = c + 1; r < 6; ++r) {
                float f = Hm[r][c] / pv;
                for (int cc = c; cc < 7; ++cc) Hm[r][cc] -= f * Hm[c][cc];
            }
        }
        float x[6];
        for (int r = 5; r >= 0; --r) {
            float v = Hm[r][6];
            for (int cc = r + 1; cc < 6; ++cc) v -= Hm[r][cc] * x[cc];
            x[r] = v / Hm[r][r];
        }
        for (int m = 0; m < M_AND; ++m)
            alpha[(size_t)b * M_AND + m] = x[1 + m];
    }
}

// z_new = sum_m alpha[b,m] * F[b,m,:]; also writes Z[:,slot] = z_new (BETA=1)
__global__ void zcombine(const float* __restrict__ alpha, const float* __restrict__ F,
                         float* __restrict__ znew, float* __restrict__ Z, int slot)
{
    int i = blockIdx.x * blockDim.x + threadIdx.x;   // 512*2048
    int b = i >> 11, d = i & 2047;
    float s = 0.f;
    #pragma unroll
    for (int m = 0; m < M_AND; ++m)
        s += alpha[b * M_AND + m] * F[((size_t)b * M_AND + m) * OUT_DIM + d];
    znew[i] = s;
    Z[((size_t)b * M_AND + slot) * OUT_DIM + d] = s;
}

// output = F[:, slot] (last single_it result == final relu(multiply+bias))
__global__ void copy_out(const float* __restrict__ F, float* __restrict__ out, int slot)
{
    int i = blockIdx.x * blockDim.x + threadIdx.x;
    int b = i >> 11, d = i & 2047;
    out[i] = F[((size_t)b * M_AND + slot) * OUT_DIM + d];
}

// ---------------------------------------------------------------------------
extern "C" void kernel_launch(void* const* d_in, const int* in_sizes, int n_in,
                              void* d_out, int out_size, void* d_ws, size_t ws_size,
                              hipStream_t stream)
{
    const float* x  = (const float*)d_in[0];   // 512 x 1024
    const float* Am = (const float*)d_in[1];   // 2048 x 2048
    const float* Bm = (const float*)d_in[2];   // 2048 x 2048
    const float* UW = (const float*)d_in[3];   // 2048 x 1024
    const float* Ub = (const float*)d_in[4];   // 2048
    float* out = (float*)d_out;

    float* ws    = (float*)d_ws;
    float* Mmat  = ws;                                     // 2048*2048
    float* bias  = Mmat + (size_t)OUT_DIM * OUT_DIM;       // 512*2048
    float* Z     = bias + (size_t)BSZ * OUT_DIM;           // 512*5*2048
    float* F     = Z + (size_t)BSZ * M_AND * OUT_DIM;      // 512*5*2048
    float* znew  = F + (size_t)BSZ * M_AND * OUT_DIM;      // 512*2048
    float* alpha = znew + (size_t)BSZ * OUT_DIM;           // 512*5

    dim3 blk(256);
    const int nElem = BSZ * OUT_DIM;

    // zero Z and F (contiguous)
    fill_zero<<<2048, blk, 0, stream>>>(Z, (long)2 * BSZ * M_AND * OUT_DIM);

    // bias = x @ U_W^T + U_b
    wmma_gemm<EPI_BIASVEC, false, true>
        <<<dim3(OUT_DIM / 64, BSZ / 128), blk, 0, stream>>>(
            x, IN_DIM, UW, IN_DIM, bias, OUT_DIM, IN_DIM, Ub, 0, 0.f);

    // Mmat = -A^T A + B^T - B + (1-m) I   (m = 1 -> no diagonal term)
    wmma_gemm<EPI_SKEW, true, false>
        <<<dim3(OUT_DIM / 64, OUT_DIM / 128), blk, 0, stream>>>(
            Am, OUT_DIM, Am, OUT_DIM, Mmat, OUT_DIM, OUT_DIM, Bm, OUT_DIM,
            1.0f - M_MON);

    // f0 = relu(bias); F[:,0] = f0, Z[:,1] = f0
    bootstrap_f0<<<nElem / 256, blk, 0, stream>>>(bias, F, Z);

    // F[:,1] = relu(f0 @ M + bias)
    wmma_gemm<EPI_RELUB, false, false>
        <<<dim3(OUT_DIM / 64, BSZ / 128), blk, 0, stream>>>(
            F + 0 * OUT_DIM, M_AND * OUT_DIM, Mmat, OUT_DIM,
            F + 1 * OUT_DIM, M_AND * OUT_DIM, OUT_DIM, bias, OUT_DIM, 0.f);

    // Anderson loop: fixed 48 iterations (it = 2 .. 49)
    for (int it = 2; it < MAX_IT; ++it) {
        int slot = it % M_AND;
        anderson_alpha<<<BSZ, blk, 0, stream>>>(Z, F, alpha, it);
        zcombine<<<nElem / 256, blk, 0, stream>>>(alpha, F, znew, Z, slot);
        wmma_gemm<EPI_RELUB, false, false>
            <<<dim3(OUT_DIM / 64, BSZ / 128), blk, 0, stream>>>(
                znew, OUT_DIM, Mmat, OUT_DIM,
                F + slot * OUT_DIM, M_AND * OUT_DIM, OUT_DIM, bias, OUT_DIM, 0.f);
    }

    copy_out<<<nElem / 256, blk, 0, stream>>>(F, out, (MAX_IT - 1) % M_AND);
}